// GNNModel_14259291423189
// MI455X (gfx1250) — compile-verified
//
#include <hip/hip_runtime.h>

// ---------------------------------------------------------------------------
// 3-layer GCN for MI455X (gfx1250, wave32).
//  - GEMMs: native fp32 WMMA (v_wmma_f32_16x16x4_f32), K fully unrolled.
//    A tile staged once per block into LDS via CDNA5 async global->LDS
//    copies (cuts A fetch traffic 8x); padded LDS stride -> conflict-free
//    ds_load_b64 A-fragment reads.
//  - Edge scatter dominates the roofline (~2.5 GB L2-resident gather +
//    f32 atomic-add traffic); float4-vectorized, one thread = 4 features.
// ---------------------------------------------------------------------------

typedef float v2f __attribute__((ext_vector_type(2)));
typedef float v8f __attribute__((ext_vector_type(8)));

// deg[i] = 1.0 (self loop)
__global__ void init_ones_k(float* __restrict__ deg, int n) {
  int i = blockIdx.x * blockDim.x + threadIdx.x;
  if (i < n) deg[i] = 1.0f;
}

// deg[dst[e]] += 1
__global__ void add_deg_k(const int* __restrict__ dst, float* __restrict__ deg, int E) {
  int e = blockIdx.x * blockDim.x + threadIdx.x;
  if (e < E) atomicAdd(&deg[dst[e]], 1.0f);
}

// dinv[i] = rsqrt(deg[i])
__global__ void make_dinv_k(const float* __restrict__ deg, float* __restrict__ dinv, int n) {
  int i = blockIdx.x * blockDim.x + threadIdx.x;
  if (i < n) dinv[i] = rsqrtf(deg[i]);
}

// One block = one 16-row tile; 8 waves cover up to NC/16 column tiles.
// A tile (16 x K fp32) is staged in LDS once per block (async copies), then
// each wave runs K/4 v_wmma_f32_16x16x4_f32 with a 16x16 fp32 accumulator.
//
// Writes BOTH:  Hout[m,n] = (X@W)[m,n]            (needed by the scatter)
//               Agg [m,n] = h*dinv[m]^2 + bias[n] (self-loop + bias init)
//
// Hout may alias X (in-place when K==NC): X is only read during LDS staging,
// which finishes (s_wait_asynccnt + barrier) before any global store, and a
// block's 16 rows are touched by no other block.
template <int K, int NC>
__global__ void __launch_bounds__(256)
gcn_gemm_wmma(const float* X, const float* __restrict__ W,
              const float* __restrict__ bias, const float* __restrict__ dinv,
              float* Hout, float* __restrict__ Agg)
{
  constexpr int PADK = K + 4;              // 528B row stride: 16B-aligned,
  __shared__ float As[16 * PADK];          // conflict-free b64 reads

  const int tid  = threadIdx.x;
  const int wave = tid >> 5;
  const int lane = tid & 31;
  const int l    = lane & 15;              // tile column / A row
  const int hi   = lane >> 4;              // half-wave select
  const int m0   = blockIdx.x << 4;

  // ---- stage A tile into LDS: 16*K/4 float4 async copies, coalesced ----
  constexpr int R4  = K / 4;               // float4 per row
  constexpr int NV4 = 16 * R4;
#pragma unroll
  for (int i = tid; i < NV4; i += 256) {
    const int row = i / R4;
    const int c4  = i % R4;
    const float* gp = X + (size_t)(m0 + row) * K + c4 * 4;
    const unsigned lds_off = (unsigned)(uintptr_t)(&As[row * PADK + c4 * 4]);
    asm volatile("global_load_async_to_lds_b128 %0, %1, off"
                 :: "v"(lds_off), "v"(gp) : "memory");
  }
  asm volatile("s_wait_asynccnt 0x0" ::: "memory");
  __syncthreads();

  v8f acc = {};
  constexpr int NTILES = NC / 16;
  if (wave < NTILES) {
    const int n0 = wave << 4;
    const float* arow = &As[l * PADK];     // A: row l of the LDS tile
    const float* bcol = W + n0 + l;        // B: column n0+l of W
#pragma unroll
    for (int k0 = 0; k0 < K; k0 += 4) {
      const int ka = k0 + (hi << 1);
      // A 16x4 fp32 layout: VGPR r = K=r (lanes 0-15) / K=r+2 (lanes 16-31)
      v2f a;
      a.x = arow[ka];
      a.y = arow[ka + 1];
      // B 4x16 fp32 layout mirrors C/D: VGPR r = row K=r / K=r+2, N = lane
      v2f b;
      b.x = bcol[(ka + 0) * NC];
      b.y = bcol[(ka + 1) * NC];
      acc = __builtin_amdgcn_wmma_f32_16x16x4_f32(
          /*neg_a=*/false, a, /*neg_b=*/false, b,
          /*c_mod=*/(short)0, acc, /*reuse_a=*/false, /*reuse_b=*/false);
    }
    const float bs = bias[n0 + l];
#pragma unroll
    for (int r = 0; r < 8; ++r) {
      // C/D layout: VGPR r holds M = r + 8*hi, N = l
      const int m  = m0 + (hi << 3) + r;
      const float di = dinv[m];
      const float h  = acc[r];
      const size_t o = (size_t)m * NC + n0 + l;
      Hout[o] = h;
      Agg[o]  = h * di * di + bs;
    }
  }
}

// agg[dst[e], :] += h[src[e], :] * dinv[src]*dinv[dst]
// One thread handles 4 features of one edge (float4 gather + 4 f32 atomics).
__global__ void gcn_scatter(const int* __restrict__ src, const int* __restrict__ dst,
                            const float* __restrict__ dinv, const float* __restrict__ h,
                            float* __restrict__ agg, long long total,
                            int c4shift, int c4mask, int Ncols)
{
  for (long long idx = (long long)blockIdx.x * blockDim.x + threadIdx.x;
       idx < total;
       idx += (long long)gridDim.x * blockDim.x) {
    const int e = (int)(idx >> c4shift);
    const int f = ((int)idx & c4mask) << 2;
    const int s = src[e];
    const int d = dst[e];
    const float nrm = dinv[s] * dinv[d];
    const float4 hv = *(const float4*)(h + (size_t)s * Ncols + f);
    float* ap = agg + (size_t)d * Ncols + f;
    atomicAdd(ap + 0, hv.x * nrm);
    atomicAdd(ap + 1, hv.y * nrm);
    atomicAdd(ap + 2, hv.z * nrm);
    atomicAdd(ap + 3, hv.w * nrm);
  }
}

// in-place relu, float4 vectorized (element counts here are multiples of 4)
__global__ void relu4_k(float* __restrict__ x, long long n4) {
  long long i = (long long)blockIdx.x * blockDim.x + threadIdx.x;
  if (i < n4) {
    float4* p = (float4*)x + i;
    float4 v = *p;
    v.x = fmaxf(v.x, 0.0f);
    v.y = fmaxf(v.y, 0.0f);
    v.z = fmaxf(v.z, 0.0f);
    v.w = fmaxf(v.w, 0.0f);
    *p = v;
  }
}

extern "C" void kernel_launch(void* const* d_in, const int* in_sizes, int n_in,
                              void* d_out, int out_size, void* d_ws, size_t ws_size,
                              hipStream_t stream) {
  const float* x  = (const float*)d_in[0];
  const int*   ei = (const int*)d_in[1];   // edge_index [2, E]
  const float* W1 = (const float*)d_in[2];
  const float* b1 = (const float*)d_in[3];
  const float* W2 = (const float*)d_in[4];
  const float* b2 = (const float*)d_in[5];
  const float* W3 = (const float*)d_in[6];
  const float* b3 = (const float*)d_in[7];
  float* out = (float*)d_out;

  const int D = 128, H = 128, C = 64;
  const int N = in_sizes[0] / D;      // 50000 (divisible by 16)
  const int E = in_sizes[1] / 2;      // 800000
  const int* srcI = ei;
  const int* dstI = ei + E;

  // workspace layout (floats): dinv[N] | deg[N] | bufA[N*H] | bufB[N*H]
  float* dinv = (float*)d_ws;
  float* deg  = dinv + N;
  float* bufA = deg + N;
  float* bufB = bufA + (size_t)N * H;

  const int T = 256;
  // degrees + symmetric norm
  init_ones_k<<<(N + T - 1) / T, T, 0, stream>>>(deg, N);
  add_deg_k  <<<(E + T - 1) / T, T, 0, stream>>>(dstI, deg, E);
  make_dinv_k<<<(N + T - 1) / T, T, 0, stream>>>(deg, dinv, N);

  const dim3 gemmGrid(N / 16);
  const long long totH = (long long)E * (H / 4);
  const long long totC = (long long)E * (C / 4);
  const int blkH = (int)((totH + T - 1) / T);
  const int blkC = (int)((totC + T - 1) / T);
  const long long n4H = (long long)N * H / 4;

  // ---- layer 1: x -> bufA (h), bufB (agg) ----
  gcn_gemm_wmma<128, 128><<<gemmGrid, T, 0, stream>>>(x, W1, b1, dinv, bufA, bufB);
  gcn_scatter<<<blkH, T, 0, stream>>>(srcI, dstI, dinv, bufA, bufB, totH, 5, 31, H);
  relu4_k<<<(int)((n4H + T - 1) / T), T, 0, stream>>>(bufB, n4H);

  // ---- layer 2: bufB -> bufB (h, in place; K==NC), bufA (agg) ----
  gcn_gemm_wmma<128, 128><<<gemmGrid, T, 0, stream>>>(bufB, W2, b2, dinv, bufB, bufA);
  gcn_scatter<<<blkH, T, 0, stream>>>(srcI, dstI, dinv, bufB, bufA, totH, 5, 31, H);
  relu4_k<<<(int)((n4H + T - 1) / T), T, 0, stream>>>(bufA, n4H);

  // ---- layer 3: bufA -> bufB (h, stride C), d_out (agg) ----
  gcn_gemm_wmma<128, 64><<<gemmGrid, T, 0, stream>>>(bufA, W3, b3, dinv, bufB, out);
  gcn_scatter<<<blkC, T, 0, stream>>>(srcI, dstI, dinv, bufB, out, totC, 4, 15, C);
}